// LoaclGeometricStructure_472446403132
// MI455X (gfx1250) — compile-verified
//
#include <hip/hip_runtime.h>

typedef __attribute__((ext_vector_type(2))) float v2f;
typedef __attribute__((ext_vector_type(8))) float v8f;

#define BDIM 256
#define WAVES_PER_BLOCK 8
#define PTS_PER_WAVE 4

// xor-shuffle across the wave32 via ds_swizzle group-of-32 encoding:
// src_lane = ((lane & 0x1f) | 0x00) ^ mask   -> offset = (mask<<10) | 0x1f
#define SWZ_XOR_F32(v, mask) \
    __int_as_float(__builtin_amdgcn_ds_swizzle(__float_as_int(v), (((mask) << 10) | 0x1f)))

// B=4, N=16384, K=16, L=32, SIGMA=0.1
// out[b,l,n] = (1/K) * sum_{k,m} exp(-(||x_k||^2 + ||kap_{l,m}||^2 - 2 x_k.kap_{l,m}) / (2 sigma^2))
// Implemented as exp2( WMMA(A,B)[k,m] + bias[k] ) with
//   A row k     = [x0, x1, x2, 1]
//   B col (l,m) = [2c*k0, 2c*k1, 2c*k2, -c*||kap||^2],  c = log2(e)/(2 sigma^2)
//   bias[k]     = -c*||x_k||^2

__global__ __launch_bounds__(BDIM) void kcnet_wmma_kernel(
    const float* __restrict__ points,   // (B,3,N)
    const int*   __restrict__ knn,      // (B,N,K)
    const float* __restrict__ kern,     // (L,K,3)
    float*       __restrict__ out,      // (B,L,N)
    int N)
{
    constexpr int K = 16, L = 32;
    constexpr float C = 72.134752044448169f;  // log2(e) / (2*0.1*0.1)

    const int lane = threadIdx.x & 31;
    const int wave = threadIdx.x >> 5;
    const int m    = lane & 15;
    const int hi   = lane >> 4;

    // ---- Preload B tiles for all 32 kernel rows (registers; loop fully unrolled) ----
    // 32-bit B 4x16 layout: VGPR0 = row K=0 (lanes 0-15) / K=2 (lanes 16-31),
    //                       VGPR1 = row K=1 (lanes 0-15) / K=3 (lanes 16-31).
    float kb0[L], kb1[L];
#pragma unroll
    for (int l = 0; l < L; ++l) {
        const float k0 = kern[(l*K + m)*3 + 0];
        const float k1 = kern[(l*K + m)*3 + 1];
        const float k2 = kern[(l*K + m)*3 + 2];
        const float kk2 = k0*k0 + k1*k1 + k2*k2;
        kb0[l] = hi ? (2.0f*C)*k2 : (2.0f*C)*k0;
        kb1[l] = hi ? (-C)*kk2    : (2.0f*C)*k1;
    }

    const long base_pt = ((long)blockIdx.x * WAVES_PER_BLOCK + wave) * PTS_PER_WAVE;

    for (int p = 0; p < PTS_PER_WAVE; ++p) {
        const long pid = base_pt + p;
        const int b = (int)(pid / N);
        const int n = (int)(pid % N);

        const float* pb = points + (long)b * 3 * N;
        const float cx0 = pb[0*N + n];
        const float cx1 = pb[1*N + n];
        const float cx2 = pb[2*N + n];

        const int nb = knn[((long)b * N + n) * K + m];
        const float x0 = pb[0*N + nb] - cx0;
        const float x1 = pb[1*N + nb] - cx1;
        const float x2 = pb[2*N + nb] - cx2;

        // A operand: 16x4 f32. Lanes 0-15: (K0,K1) = (x0,x1); lanes 16-31: (K2,K3) = (x2,1).
        v2f a;
        a.x = hi ? x2   : x0;
        a.y = hi ? 1.0f : x1;

        // ||x_k||^2 for neighbor (lane&15): combine the two half-lane partials.
        float part16 = hi ? (x2*x2) : (x0*x0 + x1*x1);
        const float x2full = part16 + SWZ_XOR_F32(part16, 16);

        // Per-C/D-row bias: row M = r + 8*hi lives at VGPR r; x2 of neighbor j is in lane j (and j+16).
        float bias[8];
#pragma unroll
        for (int r = 0; r < 8; ++r)
            bias[r] = -C * __shfl(x2full, r + 8*hi, 32);

        // ---- Main compute: one WMMA + 8 exp per kernel row; keep only per-lane row sums ----
        float part[L];
#pragma unroll
        for (int l = 0; l < L; ++l) {
            v2f bt;
            bt.x = kb0[l];
            bt.y = kb1[l];
            v8f acc = {0.f, 0.f, 0.f, 0.f, 0.f, 0.f, 0.f, 0.f};
            v8f d = __builtin_amdgcn_wmma_f32_16x16x4_f32(
                /*neg_a=*/false, a, /*neg_b=*/false, bt,
                /*c_mod=*/(short)0, acc, /*reuse_a=*/false, /*reuse_b=*/false);

            // 8 independent exps, then a pairwise add tree (depth 3) so the adds can
            // fill the TRANS32 hazard slots instead of v_nop after every v_exp_f32.
            float e[8];
#pragma unroll
            for (int r = 0; r < 8; ++r)
                e[r] = __builtin_amdgcn_exp2f(d[r] + bias[r]);
            const float s01 = e[0] + e[1];
            const float s23 = e[2] + e[3];
            const float s45 = e[4] + e[5];
            const float s67 = e[6] + e[7];
            part[l] = (s01 + s23) + (s45 + s67);   // lane j: column-group partial of tile l
        }

        // ---- Batched transpose-reduce: out[l] = sum_j part[l][lane j], result lands in lane l.
        // 5 butterfly stages, halving the per-lane array each stage; shuffles within a
        // stage are independent so their latency overlaps (31 ds ops total).
#pragma unroll
        for (int st = 0; st < 5; ++st) {
            const int d   = 16 >> st;       // 16, 8, 4, 2, 1
            const int hsz = 16 >> st;       // half-size of current array
            const bool bit = (lane & d) != 0;
#pragma unroll
            for (int i = 0; i < 16; ++i) {
                if (i < hsz) {
                    const float lo = part[i];
                    const float hic = part[i + hsz];
                    const float send = bit ? lo : hic;
                    float recv;
                    switch (d) {
                        case 16: recv = SWZ_XOR_F32(send, 16); break;
                        case  8: recv = SWZ_XOR_F32(send,  8); break;
                        case  4: recv = SWZ_XOR_F32(send,  4); break;
                        case  2: recv = SWZ_XOR_F32(send,  2); break;
                        default: recv = SWZ_XOR_F32(send,  1); break;
                    }
                    part[i] = (bit ? hic : lo) + recv;
                }
            }
        }
        // lane j now holds the full tile sum for l = j.
        out[((long)b * L + lane) * N + n] = part[0] * (1.0f / 16.0f);
    }
}

extern "C" void kernel_launch(void* const* d_in, const int* in_sizes, int n_in,
                              void* d_out, int out_size, void* d_ws, size_t ws_size,
                              hipStream_t stream) {
    const float* points = (const float*)d_in[0];   // (B,3,N) f32
    const int*   knn    = (const int*)d_in[1];     // (B,N,K) i32
    const float* kern   = (const float*)d_in[2];   // (L,K,3) f32
    float*       out    = (float*)d_out;           // (B,L,N) f32

    const int B = 4, N = 16384;
    const int total = B * N;                                   // 65536 points
    const int pts_per_block = WAVES_PER_BLOCK * PTS_PER_WAVE;  // 32
    const int blocks = total / pts_per_block;                  // 2048

    kcnet_wmma_kernel<<<blocks, BDIM, 0, stream>>>(points, knn, kern, out, N);
}